// Router_71433896067262
// MI455X (gfx1250) — compile-verified
//
#include <hip/hip_runtime.h>
#include <math.h>

typedef __attribute__((ext_vector_type(2))) float v2f;
typedef __attribute__((ext_vector_type(8))) float v8f;

#define D_DIM 512
#define H_DIM 256
#define E_NUM 10
#define BR    128    // rows per block
#define BK    64     // K-chunk width staged in LDS
#define XPAD  68     // row pitch in floats: 16B aligned, 4-bank rotation/row
#define NCOL  32     // padded output columns (10 gate | 6 zero | 10 sim | 6 zero)

// workspace layout (float offsets)
#define WS_CG    0               // 512 x 32 combined B matrix
#define WS_CS    (D_DIM * NCOL)  // per-expert scaled bias term
#define WS_WE    (WS_CS + 16)    // trust * staleness
#define WS_INVN  (WS_WE + 16)    // 1/max(||expert_feat||, eps)

// ---------- prep0: tiny per-expert scalars ----------
__global__ void prep0_kernel(const float* __restrict__ bproj,
                             const float* __restrict__ emb,
                             const float* __restrict__ ef,
                             const float* __restrict__ trust,
                             const float* __restrict__ dt,
                             float* __restrict__ ws) {
  int e = threadIdx.x;
  if (e < E_NUM) {
    float ss = 0.f;
    for (int d = 0; d < D_DIM; ++d) { float v = ef[e * D_DIM + d]; ss += v * v; }
    ws[WS_INVN + e] = 1.0f / fmaxf(sqrtf(ss), 1e-8f);
    float cb = 0.f;
    for (int h = 0; h < H_DIM; ++h) cb += bproj[h] * emb[e * H_DIM + h];
    ws[WS_CS + e] = cb * 0.0625f;                 // * 1/sqrt(256)
    ws[WS_WE + e] = trust[e] * __expf(-0.001f * dt[e]);
  }
}

// ---------- prep1: build combined 512x32 B matrix ----------
__global__ void prep1_kernel(const float* __restrict__ W,
                             const float* __restrict__ emb,
                             const float* __restrict__ ef,
                             float* __restrict__ ws) {
  int tid = blockIdx.x * blockDim.x + threadIdx.x;   // 0 .. 5119
  if (tid >= D_DIM * E_NUM) return;
  int d = tid / E_NUM, e = tid % E_NUM;
  float s = 0.f;
  for (int h = 0; h < H_DIM; ++h) s += W[d * H_DIM + h] * emb[e * H_DIM + h];
  float* Cg = ws + WS_CG;
  Cg[d * NCOL + e]      = s * 0.0625f;                      // gate-logit column
  Cg[d * NCOL + 16 + e] = ef[e * D_DIM + d] * ws[WS_INVN + e]; // normalized expert col
  if (e == 0) { for (int c = E_NUM; c < 16; ++c)      Cg[d * NCOL + c] = 0.f; }
  if (e == 1) { for (int c = 16 + E_NUM; c < 32; ++c) Cg[d * NCOL + c] = 0.f; }
}

// ---------- main fused router ----------
__global__ __launch_bounds__(256)
void router_main_kernel(const float* __restrict__ X,
                        const float* __restrict__ ws,
                        float* __restrict__ out, int B) {
  __shared__ float Xs[BR][XPAD];
  __shared__ float norm2[BR];

  const float* __restrict__ Cg = ws + WS_CG;
  const int tid  = threadIdx.x;
  const int lane = tid & 31;
  const int wave = tid >> 5;
  const int half = lane >> 4;     // 0: K+0/K+1, 1: K+2/K+3 (A & B frag layout)
  const int l16  = lane & 15;
  const int r0   = wave * 16;     // this wave owns 16 rows
  const long rowBase = (long)blockIdx.x * BR;

  if (tid < BR) norm2[tid] = 0.f;

  const int lrow16 = tid >> 4;    // 0..15 : row-within-pass for loads
  const int c4     = tid & 15;    // float4 column within row chunk
  float sq[8];
  #pragma unroll
  for (int p = 0; p < 8; ++p) sq[p] = 0.f;

  v8f acc0 = {};   // columns 0..15  (gate dots)
  v8f acc1 = {};   // columns 16..31 (sim dots)

  for (int c = 0; c < D_DIM / BK; ++c) {
    __syncthreads();
    // cooperative coalesced load of 128 x 64 chunk + private ||x||^2 partials
    #pragma unroll
    for (int p = 0; p < 8; ++p) {
      int row = p * 16 + lrow16;
      const float4 v = *reinterpret_cast<const float4*>(
          &X[(rowBase + row) * D_DIM + c * BK + c4 * 4]);
      *reinterpret_cast<float4*>(&Xs[row][c4 * 4]) = v;
      sq[p] += v.x * v.x + v.y * v.y + v.z * v.z + v.w * v.w;
    }
    __syncthreads();

    const float* xrow = &Xs[r0 + l16][0];
    #pragma unroll
    for (int kk = 0; kk < BK / 4; ++kk) {
      int k  = kk * 4 + half * 2;           // local K within chunk
      v2f a  = *reinterpret_cast<const v2f*>(&xrow[k]);   // A frag: 16x4 f32 layout
      int kg = c * BK + k;                  // global K for B matrix
      v2f b0, b1;
      b0.x = Cg[kg * NCOL + l16];
      b0.y = Cg[(kg + 1) * NCOL + l16];
      b1.x = Cg[kg * NCOL + 16 + l16];
      b1.y = Cg[(kg + 1) * NCOL + 16 + l16];
      acc0 = __builtin_amdgcn_wmma_f32_16x16x4_f32(
          false, a, false, b0, (short)0, acc0, false, false);
      acc1 = __builtin_amdgcn_wmma_f32_16x16x4_f32(
          false, a, false, b1, (short)0, acc1, false, false);
    }
  }

  // fold private norm partials into per-row LDS accumulators
  #pragma unroll
  for (int p = 0; p < 8; ++p) atomicAdd(&norm2[p * 16 + lrow16], sq[p]);

  __syncthreads();   // all waves done reading Xs -> safe to reuse as scratch
  float* S = &Xs[0][0];    // S[row*33 + col], 128*33 floats fit inside Xs
  #pragma unroll
  for (int j = 0; j < 8; ++j) {
    int row = r0 + half * 8 + j;   // C/D layout: VGPR j -> M=j (lanes 0-15), M=j+8 (16-31)
    S[row * 33 + l16]      = acc0[j];
    S[row * 33 + 16 + l16] = acc1[j];
  }
  __syncthreads();

  if (tid < BR) {
    const int  row  = tid;
    const long grow = rowBase + row;
    const float invx = 1.0f / fmaxf(sqrtf(norm2[row]), 1e-8f);
    const float* cs = ws + WS_CS;
    const float* we = ws + WS_WE;

    float sc[E_NUM];
    float mx = -1e30f;
    #pragma unroll
    for (int e = 0; e < E_NUM; ++e) {
      float gl   = S[row * 33 + e] + cs[e];
      float gate = 1.0f / (1.0f + __expf(-gl));
      float sim  = (S[row * 33 + 16 + e] * invx + 1.0f) * 0.5f;
      float v    = gate * we[e] * sim;
      sc[e] = v;
      mx = fmaxf(mx, v);
    }
    float pr[E_NUM], sum = 0.f;
    #pragma unroll
    for (int e = 0; e < E_NUM; ++e) { pr[e] = __expf(sc[e] - mx); sum += pr[e]; }
    float isum = 1.0f / sum;
    #pragma unroll
    for (int e = 0; e < E_NUM; ++e) pr[e] *= isum;

    // top-3 (ties -> lowest index first, matching jax.lax.top_k)
    float tmp[E_NUM];
    #pragma unroll
    for (int e = 0; e < E_NUM; ++e) tmp[e] = pr[e];
    float tw[3]; int ti[3];
    #pragma unroll
    for (int k = 0; k < 3; ++k) {
      float bv = -1.0f; int bi = 0;
      #pragma unroll
      for (int e = 0; e < E_NUM; ++e) { if (tmp[e] > bv) { bv = tmp[e]; bi = e; } }
      tw[k] = bv; ti[k] = bi; tmp[bi] = -2.0f;
    }
    float inv3 = 1.0f / (tw[0] + tw[1] + tw[2]);

    // outputs: [topk_w (B,3)][topk_idx (B,3) int bits][probs (B,10)]
    int* iout = reinterpret_cast<int*>(out);
    #pragma unroll
    for (int k = 0; k < 3; ++k) {
      out[grow * 3 + k]                 = tw[k] * inv3;
      iout[(long)B * 3 + grow * 3 + k]  = ti[k];
    }
    #pragma unroll
    for (int e = 0; e < E_NUM; ++e)
      out[(long)B * 6 + grow * 10 + e] = pr[e];
  }
}

extern "C" void kernel_launch(void* const* d_in, const int* in_sizes, int n_in,
                              void* d_out, int out_size, void* d_ws, size_t ws_size,
                              hipStream_t stream) {
  const float* features = (const float*)d_in[0];
  const float* W_proj   = (const float*)d_in[1];
  const float* b_proj   = (const float*)d_in[2];
  const float* emb      = (const float*)d_in[3];
  const float* ef       = (const float*)d_in[4];
  const float* trust    = (const float*)d_in[5];
  const float* dt       = (const float*)d_in[6];
  float* ws  = (float*)d_ws;
  float* out = (float*)d_out;
  const int B = in_sizes[0] / D_DIM;

  prep0_kernel<<<1, 64, 0, stream>>>(b_proj, emb, ef, trust, dt, ws);
  prep1_kernel<<<(D_DIM * E_NUM + 255) / 256, 256, 0, stream>>>(W_proj, emb, ef, ws);
  router_main_kernel<<<B / BR, 256, 0, stream>>>(features, ws, out, B);
}